// QuantumRegressionModel_65481071402886
// MI455X (gfx1250) — compile-verified
//
#include <hip/hip_runtime.h>
#include <hip/hip_bf16.h>
#include <math.h>

// MI455X (gfx1250) implementation of a 16-qubit variational circuit + linear head.
//
// Roofline: state = 64 x 65536 c64 = 32MB (L2-resident, 192MB L2). Total math
// ~3.2 GFLOP -> compute-trivial; bound by 12 in-place L2 passes (~768MB traffic).
// Gates are refactored into 16x16 complex matrices applied per nibble axis via
// chained V_WMMA_F32_16X16X4_F32 (4 k-chunks x 4 real matmuls = 16 wmma / tile).
// Pass strides are template constants so all row offsets fold into the 24-bit
// global_load/store immediates (one 64-bit base add per lane instead of chains).

typedef __attribute__((ext_vector_type(2))) float v2f;
typedef __attribute__((ext_vector_type(8))) float v8f;

#define DIMQ 65536
#define BATCHN 64

struct c32 { float re, im; };
__device__ __forceinline__ c32 cmul(c32 a, c32 b) {
  c32 r; r.re = a.re * b.re - a.im * b.im; r.im = a.re * b.im + a.im * b.re; return r;
}

// ---------------- kernel: copy input state into mutable workspace ----------------
__global__ void copy_state(const float* __restrict__ sr, const float* __restrict__ si,
                           float* __restrict__ pr, float* __restrict__ pi, int n) {
  int i = blockIdx.x * blockDim.x + threadIdx.x;
  int stride = gridDim.x * blockDim.x;
  for (; i < n; i += stride) { pr[i] = sr[i]; pi[i] = si[i]; }
}

// ---------------- kernel: build 16x16 group matrices ----------------
// For layer l, group g (qubits 4g..4g+3), sel in {0,1}:
//   R = U_{4g} (x) U_{4g+1} (x) U_{4g+2} (x) U_{4g+3}   (rotations, wire 4g = MSB)
//   P = intra-group CNOT chain permutation: b2^=b3; b1^=b2; b0^=b1
//   E (sel=0) = P * R            (incoming cross-CNOT control bit = 0)
//   O (sel=1) = P * X_msb * R    (control bit = 1: flip qubit 4g, the group MSB)
// Layout: W[(l*4+g)*1024 + sel*512 + comp*256 + j*16 + k], comp 0=re, 1=im.
__global__ void __launch_bounds__(384) build_mats(const float* __restrict__ vp,
                                                  float* __restrict__ W) {
  int tid = threadIdx.x;           // 384 = 3 layers * 4 groups * 2 sel * 16 rows
  if (tid >= 384) return;
  int l   = tid >> 7;
  int g   = (tid >> 5) & 3;
  int sel = (tid >> 4) & 1;
  int j   = tid & 15;

  // Single-qubit U = Rz(tz) Ry(ty) Rx(tx) for the 4 qubits of this group.
  c32 U[4][2][2];
  for (int t = 0; t < 4; ++t) {
    int q = 4 * g + t;
    const float* p = vp + (size_t)(l * 16 + q) * 6;   // N_PARAMS=6, use [0..2]
    float cx = cosf(0.5f * p[0]), sx = sinf(0.5f * p[0]);
    float cy = cosf(0.5f * p[1]), sy = sinf(0.5f * p[1]);
    float cz = cosf(0.5f * p[2]), sz = sinf(0.5f * p[2]);
    // M1 = Ry @ Rx
    c32 m00 = {  cy * cx,  sy * sx };
    c32 m01 = { -sy * cx, -cy * sx };
    c32 m10 = {  sy * cx, -cy * sx };
    c32 m11 = {  cy * cx, -sy * sx };
    c32 e0 = { cz, -sz }, e1 = { cz, sz };            // Rz diagonal
    U[t][0][0] = cmul(e0, m00); U[t][0][1] = cmul(e0, m01);
    U[t][1][0] = cmul(e1, m10); U[t][1][1] = cmul(e1, m11);
  }

  int trow = sel ? (j ^ 8) : j;     // X_msb folded in for sel=1
  int b3 = (j >> 3) & 1;
  int b2 = ((j >> 2) & 1) ^ b3;
  int b1 = ((j >> 1) & 1) ^ b2;
  int b0 = (j & 1) ^ b1;
  int orow = (b3 << 3) | (b2 << 2) | (b1 << 1) | b0;  // P: row j of R -> row pi(j)

  int mslot = (l * 4 + g) * 2 + sel;
  for (int k = 0; k < 16; ++k) {
    c32 v = { 1.f, 0.f };
    for (int t = 0; t < 4; ++t) {
      int bj = (trow >> (3 - t)) & 1;
      int bk = (k    >> (3 - t)) & 1;
      v = cmul(v, U[t][bj][bk]);
    }
    W[((mslot * 2 + 0) * 16 + orow) * 16 + k] = v.re;
    W[((mslot * 2 + 1) * 16 + orow) * 16 + k] = v.im;
  }
}

// ---------------- kernel: apply one 16x16 complex matrix along one nibble axis ----
// One wave per 16x16 tile: D = M * X, complex, via 16x V_WMMA_F32_16X16X4_F32.
// A-operand (32-bit 16x4, per ISA): lane<16 -> row M=lane, K pair {4kc,4kc+1};
// lane>=16 -> row M=lane-16, K pair {4kc+2,4kc+3}. B mirrors the half-wave K
// split; C/D per documented 16x16 f32 layout (VGPR v: rows v / v+8).
// RS/H0/H1/CS/SELBIT are compile-time so row offsets become load immediates.
template<int RS, int H0, int H1, int CS, int SELBIT>
__global__ void __launch_bounds__(256)
apply_group(float* __restrict__ pr, float* __restrict__ pi,
            const float* __restrict__ mats) {  // [sel][2][16][16] for this (l,g)
  unsigned wave = (blockIdx.x << 3) + (threadIdx.x >> 5);
  unsigned lane = threadIdx.x & 31;
  unsigned b = wave >> 8;          // batch element
  unsigned t = wave & 255;         // tile id within batch
  unsigned u = t >> 4, v = t & 15; // the two non-K, non-column nibble axes
  unsigned col  = lane & 15;
  unsigned half = lane >> 4;

  unsigned ioff = u * H0 + v * H1;                       // intra-batch base offset
  unsigned sel  = (SELBIT >= 0) ? ((ioff >> SELBIT) & 1) // cross-CNOT control bit
                                : 0u;                    // (wave-uniform)
  const float* M = mats + sel * 512;
  unsigned base = b * DIMQ + ioff + col * CS;
  float* prb = pr + base;
  float* pib = pi + base;

  v2f ar[4], ai[4], an[4], xr[4], xi[4];
  unsigned row = col;                          // A-operand row index = lane&15
  #pragma unroll
  for (int kc = 0; kc < 4; ++kc) {
    unsigned k0 = kc * 4 + half * 2;
    ar[kc].x = M[row * 16 + k0];          ar[kc].y = M[row * 16 + k0 + 1];
    ai[kc].x = M[256 + row * 16 + k0];    ai[kc].y = M[256 + row * 16 + k0 + 1];
    an[kc].x = -ai[kc].x;                 an[kc].y = -ai[kc].y; // f32 WMMA: no A-neg
  }
  // X loads: offsets k*RS are compile-time per unrolled step -> load immediates.
  #pragma unroll
  for (int kc = 0; kc < 4; ++kc) {
    #pragma unroll
    for (int vgpr = 0; vgpr < 2; ++vgpr) {
      unsigned k = kc * 4 + half * 2 + vgpr;
      xr[kc][vgpr] = prb[(size_t)k * RS];
      xi[kc][vgpr] = pib[(size_t)k * RS];
    }
  }

  v8f dr = {}; v8f di = {};
  #pragma unroll
  for (int kc = 0; kc < 4; ++kc) {
    // Dr = Mr*Xr + (-Mi)*Xi ; Di = Mi*Xr + Mr*Xi
    dr = __builtin_amdgcn_wmma_f32_16x16x4_f32(false, ar[kc], false, xr[kc], (short)0, dr, false, false);
    dr = __builtin_amdgcn_wmma_f32_16x16x4_f32(false, an[kc], false, xi[kc], (short)0, dr, false, false);
    di = __builtin_amdgcn_wmma_f32_16x16x4_f32(false, ai[kc], false, xr[kc], (short)0, di, false, false);
    di = __builtin_amdgcn_wmma_f32_16x16x4_f32(false, ar[kc], false, xi[kc], (short)0, di, false, false);
  }

  #pragma unroll
  for (int vv = 0; vv < 8; ++vv) {
    unsigned jrow = vv + half * 8;             // documented C/D row mapping
    prb[(size_t)jrow * RS] = dr[vv];
    pib[(size_t)jrow * RS] = di[vv];
  }
}

// ---------------- kernel: fused probs -> PauliZ expvals -> linear head -----------
// out[b] = sum_idx |psi|^2 * coef(idx) + bias, coef(idx) = sum_q (1-2*bit_{15-q}) w_q
__global__ void __launch_bounds__(256)
head_reduce(const float* __restrict__ pr, const float* __restrict__ pi,
            const float* __restrict__ hw, const float* __restrict__ hb,
            float* __restrict__ out) {
  __shared__ float w[16];
  __shared__ float red[256];
  int b = blockIdx.x;
  int tid = threadIdx.x;
  if (tid < 16) w[tid] = hw[tid];
  __syncthreads();

  const float* r  = pr + (size_t)b * DIMQ;
  const float* im = pi + (size_t)b * DIMQ;
  float acc = 0.f;
  for (int idx = tid; idx < DIMQ; idx += 256) {
    float p = r[idx] * r[idx] + im[idx] * im[idx];
    float coef = 0.f;
    #pragma unroll
    for (int q = 0; q < 16; ++q)
      coef += ((idx >> (15 - q)) & 1) ? -w[q] : w[q];
    acc += p * coef;
  }
  red[tid] = acc;
  __syncthreads();
  for (int s = 128; s > 0; s >>= 1) {
    if (tid < s) red[tid] += red[tid + s];
    __syncthreads();
  }
  if (tid == 0) out[b] = red[0] + hb[0];
}

extern "C" void kernel_launch(void* const* d_in, const int* in_sizes, int n_in,
                              void* d_out, int out_size, void* d_ws, size_t ws_size,
                              hipStream_t stream) {
  (void)in_sizes; (void)n_in; (void)out_size; (void)ws_size;
  const float* sr = (const float*)d_in[0];
  const float* si = (const float*)d_in[1];
  const float* vp = (const float*)d_in[2];   // var_params [3][16][6]
  const float* hw = (const float*)d_in[3];   // head_w [1][16]
  const float* hb = (const float*)d_in[4];   // head_b [1]
  float* out = (float*)d_out;

  float* pr = (float*)d_ws;                          // 64*65536 f32 = 16MB
  float* pi = pr + (size_t)BATCHN * DIMQ;            // 16MB
  float* W  = pi + (size_t)BATCHN * DIMQ;            // 48KB of matrices

  int n = BATCHN * DIMQ;
  copy_state<<<2048, 256, 0, stream>>>(sr, si, pr, pi, n);
  build_mats<<<1, 384, 0, stream>>>(vp, W);

  // Pass g applies along nibble axis g (stride RS); columns from the axis with
  // stride CS; u/v enumerate the remaining two axes (strides H0/H1). SELBIT
  // picks the incoming cross-CNOT control bit (LSB of nibble g-1); -1 = none.
  for (int l = 0; l < 3; ++l) {
    const float* Wl = W + (size_t)l * 4096;
    apply_group<4096,  256,  16,   1, -1><<<2048, 256, 0, stream>>>(pr, pi, Wl + 0 * 1024);
    apply_group< 256, 4096,  16,   1, 12><<<2048, 256, 0, stream>>>(pr, pi, Wl + 1 * 1024);
    apply_group<  16, 4096, 256,   1,  8><<<2048, 256, 0, stream>>>(pr, pi, Wl + 2 * 1024);
    apply_group<   1, 4096,  16, 256,  4><<<2048, 256, 0, stream>>>(pr, pi, Wl + 3 * 1024);
  }

  head_reduce<<<BATCHN, 256, 0, stream>>>(pr, pi, hw, hb, out);
}